// GConvLSTMCell_55877524521590
// MI455X (gfx1250) — compile-verified
//
#include <hip/hip_runtime.h>
#include <hip/hip_bf16.h>

#define HID 128

typedef __attribute__((ext_vector_type(2))) float v2f;
typedef __attribute__((ext_vector_type(8))) float v8f;

// ---------------- degree kernels ----------------
__global__ void deg_init_kernel(float* __restrict__ deg, long long n) {
    long long i = (long long)blockIdx.x * blockDim.x + threadIdx.x;
    if (i < n) deg[i] = 1.0f;   // self loop contributes 1
}

__global__ void deg_count_kernel(const long long* __restrict__ dst,
                                 float* __restrict__ deg, long long E) {
    long long e = (long long)blockIdx.x * blockDim.x + threadIdx.x;
    if (e < E) atomicAdd(&deg[dst[e]], 1.0f);
}

__global__ void deg_rsqrt_kernel(float* __restrict__ deg, long long n) {
    long long i = (long long)blockIdx.x * blockDim.x + threadIdx.x;
    if (i < n) deg[i] = __frsqrt_rn(deg[i]);
}

// ---------------- GEMM: h = (x + h_cur) @ W[:, :128] via V_WMMA_F32_16X16X4_F32 --
// W is (128, 512) row-major; we use columns [0,128).
__global__ __launch_bounds__(256) void gemm_kernel(const float* __restrict__ x,
                                                   const float* __restrict__ hc,
                                                   const float* __restrict__ W,
                                                   float* __restrict__ hout,
                                                   long long n) {
    __shared__ float Wl[HID * HID];  // 64 KB: W[:, :128]
    int t = threadIdx.x;
    // cooperative staging, float4 per thread per iter
    for (int idx = t; idx < (HID * HID) / 4; idx += 256) {
        int r = idx >> 5;             // row 0..127
        int c4 = (idx & 31) << 2;     // col 0,4,...,124
        float4 v = *(const float4*)(W + (long long)r * 512 + c4);
        *(float4*)(Wl + r * HID + c4) = v;
    }
    __syncthreads();

    int wave = t >> 5;
    int lane = t & 31;
    long long rowTile = (long long)blockIdx.x * 8 + wave;
    long long rowBase = rowTile * 16;
    if (rowBase >= n) return;  // wave-uniform exit; EXEC stays all-ones for WMMA

    int m   = lane & 15;          // A-matrix: M = lane%16
    int kh  = (lane >> 4) << 1;   // A/B: lanes 16-31 hold K+2,K+3
    int nco = lane & 15;          // B/C/D: N = lane%16

    long long row = rowBase + m;
    const float* xr = x  + row * HID;
    const float* hr = hc + row * HID;

    v8f acc[8] = {};
    #pragma unroll 4
    for (int kk = 0; kk < 32; ++kk) {
        int k0 = (kk << 2) + kh;
        v2f a;
        a.x = xr[k0]     + hr[k0];
        a.y = xr[k0 + 1] + hr[k0 + 1];
        #pragma unroll
        for (int tc = 0; tc < 8; ++tc) {
            v2f bf;
            bf.x = Wl[k0 * HID + tc * 16 + nco];
            bf.y = Wl[(k0 + 1) * HID + tc * 16 + nco];
            acc[tc] = __builtin_amdgcn_wmma_f32_16x16x4_f32(
                false, a, false, bf, (short)0, acc[tc], false, false);
        }
    }

    // C/D layout: VGPR r holds row M=r (lanes 0-15) / M=r+8 (lanes 16-31), N=lane%16
    int half = lane >> 4;
    #pragma unroll
    for (int tc = 0; tc < 8; ++tc) {
        #pragma unroll
        for (int r = 0; r < 8; ++r) {
            long long orow = rowBase + r + 8 * half;
            hout[orow * HID + tc * 16 + nco] = acc[tc][r];
        }
    }
}

// ---------------- out = b + dinv^2 * h (self loop + bias) ----------------
__global__ void self_bias_kernel(const float* __restrict__ h,
                                 const float* __restrict__ dinv,
                                 const float* __restrict__ b,
                                 float* __restrict__ out, long long total) {
    long long i = (long long)blockIdx.x * blockDim.x + threadIdx.x;
    if (i < total) {
        long long row = i >> 7;
        int col = (int)(i & 127);
        float di = dinv[row];
        out[i] = b[col] + di * di * h[i];
    }
}

// ---------------- edge scatter: out[dst] += dinv[s]*dinv[d]*h[src] -------
// one wave per edge; lane owns a float4 (128 cols / 32 lanes)
__global__ void scatter_kernel(const long long* __restrict__ src,
                               const long long* __restrict__ dst,
                               const float* __restrict__ dinv,
                               const float* __restrict__ h,
                               float* __restrict__ out, long long E) {
    long long idx = (long long)blockIdx.x * blockDim.x + threadIdx.x;
    long long e = idx >> 5;
    int lane = (int)(idx & 31);
    if (e >= E) return;
    long long s = src[e];
    long long d = dst[e];
    float w = dinv[s] * dinv[d];
    float4 v = *(const float4*)(h + s * HID + lane * 4);
    float* od = out + d * HID + lane * 4;
    atomicAdd(od + 0, w * v.x);
    atomicAdd(od + 1, w * v.y);
    atomicAdd(od + 2, w * v.z);
    atomicAdd(od + 3, w * v.w);
}

extern "C" void kernel_launch(void* const* d_in, const int* in_sizes, int n_in,
                              void* d_out, int out_size, void* d_ws, size_t ws_size,
                              hipStream_t stream) {
    const float*     x     = (const float*)d_in[0];
    const long long* ei    = (const long long*)d_in[1];
    const float*     h_cur = (const float*)d_in[2];
    // d_in[3] = c_cur (unused by the sliced output)
    const float*     W     = (const float*)d_in[4];
    const float*     b     = (const float*)d_in[5];
    float*           out   = (float*)d_out;

    long long N = (long long)in_sizes[0] / HID;
    long long E = (long long)in_sizes[1] / 2;
    const long long* src = ei;
    const long long* dst = ei + E;

    // workspace: [deg/dinv: N floats][h: N*128 floats]
    float* deg = (float*)d_ws;
    size_t degBytes = (((size_t)N * sizeof(float)) + 255) & ~(size_t)255;
    float* h = (float*)((char*)d_ws + degBytes);

    const int B = 256;

    deg_init_kernel<<<(unsigned)((N + B - 1) / B), B, 0, stream>>>(deg, N);
    deg_count_kernel<<<(unsigned)((E + B - 1) / B), B, 0, stream>>>(dst, deg, E);
    deg_rsqrt_kernel<<<(unsigned)((N + B - 1) / B), B, 0, stream>>>(deg, N);

    long long tiles = (N + 15) / 16;
    unsigned gemmBlocks = (unsigned)((tiles + 7) / 8);
    gemm_kernel<<<gemmBlocks, B, 0, stream>>>(x, h_cur, W, h, N);

    long long total = N * HID;
    self_bias_kernel<<<(unsigned)((total + B - 1) / B), B, 0, stream>>>(h, deg, b, out, total);

    long long sthreads = E * 32;
    scatter_kernel<<<(unsigned)((sthreads + B - 1) / B), B, 0, stream>>>(src, dst, deg, h, out, E);
}